// FlowGNN_59442347377020
// MI455X (gfx1250) — compile-verified
//
#include <hip/hip_runtime.h>
#include <cmath>

typedef float v2f __attribute__((ext_vector_type(2)));
typedef float v8f __attribute__((ext_vector_type(8)));

#define NN   50000
#define DNB  32
#define BB   1024
#define CH   128
#define M1   (BB*DNB + BB)   /* 33792 */
#define EPSV 1e-5f

// ---------------------------------------------------------------------------
// Kernel 1: build layer-1 node list cn[] and timestamps cts[]
//   cn = concat(edge_idx[batch].flatten(), batch)
//   cts = concat(edge_ts[batch].flatten(), node_ts[batch])
// ---------------------------------------------------------------------------
__global__ void build_cn_cts(const int* __restrict__ batch,
                             const int* __restrict__ edge_idx,
                             const float* __restrict__ edge_ts,
                             const float* __restrict__ node_ts,
                             int* __restrict__ cn, float* __restrict__ cts) {
  int tid = blockIdx.x * blockDim.x + threadIdx.x;
  if (tid < BB * DNB) {
    int i = tid >> 5, j = tid & 31;
    int bu = batch[i];
    cn[tid]  = edge_idx[(size_t)bu * DNB + j];
    cts[tid] = edge_ts[(size_t)bu * DNB + j];
  } else if (tid < M1) {
    int i  = tid - BB * DNB;
    int bu = batch[i];
    cn[tid]  = bu;
    cts[tid] = node_ts[bu];
  }
}

// Wave-wide OR-reduce of a per-lane bit to build a 32-bit membership mask
// (avoids relying on __ballot's width on gfx1250 wave32).
__device__ __forceinline__ unsigned wave_mask(bool pred, int lane) {
  unsigned m = pred ? (1u << lane) : 0u;
  #pragma unroll
  for (int off = 16; off; off >>= 1) m |= (unsigned)__shfl_xor((int)m, off, 32);
  return m;
}

// ---------------------------------------------------------------------------
// Kernel 2: layer-1 masked mean aggregation. One wave per node m in [0,M1).
//   agg[m] = (sum_{j in mask} x[edge_idx[u][j]] + x[u]) / (|mask|+1)
//   mask j: (j < degrees[u]) && (edge_ts[u][j] <= cts[m]),  u = cn[m]
// Lane l owns channels 4l..4l+3 (float4, coalesced 512B per row).
// ---------------------------------------------------------------------------
__global__ void __launch_bounds__(256)
agg_layer1(const float* __restrict__ x,
           const float* __restrict__ edge_ts,
           const int* __restrict__ edge_idx,
           const int* __restrict__ degrees,
           const int* __restrict__ cn,
           const float* __restrict__ cts,
           float* __restrict__ agg) {
  int lane = threadIdx.x & 31;
  int m = blockIdx.x * (blockDim.x >> 5) + (threadIdx.x >> 5);
  int u = __builtin_amdgcn_readfirstlane(cn[m]);        // wave-uniform node id
  float t = cts[m];                                     // uniform value per wave
  int deg = __builtin_amdgcn_readfirstlane(degrees[u]);

  float ets = edge_ts[(size_t)u * DNB + lane];          // coalesced 128B
  int  vidx = edge_idx[(size_t)u * DNB + lane];         // coalesced 128B

  unsigned mask = wave_mask((lane < deg) && (ets <= t), lane);
  float inv = 1.0f / (float)(__popc(mask) + 1);

  const float4* xs = (const float4*)(x + (size_t)u * CH);
  float4 acc = xs[lane];                                // self embedding
  for (unsigned mm = mask; mm; mm &= mm - 1) {
    int j = __ffs(mm) - 1;                              // wave-uniform
    int v = __shfl(vidx, j, 32);                        // broadcast neighbor id
    const float4* xr = (const float4*)(x + (size_t)v * CH);
    float4 r = xr[lane];
    acc.x += r.x; acc.y += r.y; acc.z += r.z; acc.w += r.w;
  }
  float4 o; o.x = acc.x * inv; o.y = acc.y * inv; o.z = acc.z * inv; o.w = acc.w * inv;
  ((float4*)(agg + (size_t)m * CH))[lane] = o;
}

// ---------------------------------------------------------------------------
// Kernel 3: fused GEMM(M x 128 @ 128 x 128) + BatchNorm + ReLU via
// V_WMMA_F32_16X16X4_F32 (full fp32 fidelity; GEMM is not the bottleneck).
// Block = 256 threads = 8 waves; block owns 16 rows, wave w owns cols 16w..16w+15.
// A tile staged in LDS (16 x 132 padded, conflict-free b64 fragment reads),
// which also makes in-place A==out safe (reads complete before any store).
// ---------------------------------------------------------------------------
__global__ void __launch_bounds__(256)
gemm_bn_relu(const float* __restrict__ A, const float* __restrict__ W,
             const float* __restrict__ bias, const float* __restrict__ gamma,
             const float* __restrict__ beta, const float* __restrict__ rmean,
             const float* __restrict__ rvar, float* __restrict__ out) {
  __shared__ float As[16 * 132];
  const int tid = threadIdx.x;
  const int rowbase = blockIdx.x * 16;

  // Stage 16x128 A tile: each thread moves 8 floats (2 x b128).
  {
    int r  = tid >> 4;              // 0..15
    int c0 = (tid & 15) * 8;        // 0,8,...,120
    const float4* src = (const float4*)(A + (size_t)(rowbase + r) * CH + c0);
    float4 a0 = src[0], a1 = src[1];
    *(float4*)(&As[r * 132 + c0])     = a0;
    *(float4*)(&As[r * 132 + c0 + 4]) = a1;
  }
  __syncthreads();

  const int lane = tid & 31;
  const int w    = tid >> 5;        // column tile 0..7
  const int half = lane >> 4;       // 0: K=0,1 / rows M=r ; 1: K=2,3 / rows M=r+8
  const int l15  = lane & 15;
  const int col  = w * 16 + l15;    // global output column
  const float* Wc = W + col;

  v8f c = {};
  #pragma unroll 4
  for (int k = 0; k < CH; k += 4) {
    int kk = k + half * 2;
    // A fragment: lane<16 -> A[l15][k],A[l15][k+1]; lane>=16 -> A[l15][k+2],A[l15][k+3]
    v2f a = *(const v2f*)(&As[l15 * 132 + kk]);
    // B fragment: lane<16 -> W[k][col],W[k+1][col]; lane>=16 -> W[k+2][col],W[k+3][col]
    v2f b;
    b.x = Wc[(size_t)kk * CH];
    b.y = Wc[(size_t)(kk + 1) * CH];
    c = __builtin_amdgcn_wmma_f32_16x16x4_f32(false, a, false, b, (short)0, c,
                                              false, false);
  }

  // BN + ReLU epilogue, per-column parameters.
  float bv  = bias[col];
  float rm  = rmean[col];
  float rv  = rvar[col];
  float gv  = gamma[col];
  float bev = beta[col];
  float scale = gv * rsqrtf(rv + EPSV);

  #pragma unroll
  for (int r = 0; r < 8; ++r) {
    float v = (c[r] + bv - rm) * scale + bev;
    v = v > 0.0f ? v : 0.0f;
    out[(size_t)(rowbase + r + half * 8) * CH + col] = v;   // C: VGPR r -> M=r / r+8
  }
}

// ---------------------------------------------------------------------------
// Kernel 4: layer-2 masked mean. One wave per batch element i.
//   agg2[i] = (sum_{j in mask} h1[i*32+j] + h1[B*32+i]) / (|mask|+1)
//   mask j: (j < degrees[batch[i]]) && (edge_ts[batch[i]][j] <= node_ts[batch[i]])
// ---------------------------------------------------------------------------
__global__ void __launch_bounds__(256)
agg_layer2(const float* __restrict__ h1,
           const float* __restrict__ edge_ts,
           const int* __restrict__ degrees,
           const int* __restrict__ batch,
           const float* __restrict__ node_ts,
           float* __restrict__ agg) {
  int lane = threadIdx.x & 31;
  int i = blockIdx.x * (blockDim.x >> 5) + (threadIdx.x >> 5);
  int bu = __builtin_amdgcn_readfirstlane(batch[i]);
  float t = node_ts[bu];
  int deg = __builtin_amdgcn_readfirstlane(degrees[bu]);

  float ets = edge_ts[(size_t)bu * DNB + lane];
  unsigned mask = wave_mask((lane < deg) && (ets <= t), lane);
  float inv = 1.0f / (float)(__popc(mask) + 1);

  const float4* hs = (const float4*)(h1 + (size_t)(BB * DNB + i) * CH);
  float4 acc = hs[lane];                                 // self
  for (unsigned mm = mask; mm; mm &= mm - 1) {
    int j = __ffs(mm) - 1;                               // wave-uniform
    const float4* hr = (const float4*)(h1 + (size_t)(i * DNB + j) * CH);
    float4 r = hr[lane];
    acc.x += r.x; acc.y += r.y; acc.z += r.z; acc.w += r.w;
  }
  float4 o; o.x = acc.x * inv; o.y = acc.y * inv; o.z = acc.z * inv; o.w = acc.w * inv;
  ((float4*)(agg + (size_t)i * CH))[lane] = o;
}

// ---------------------------------------------------------------------------
extern "C" void kernel_launch(void* const* d_in, const int* in_sizes, int n_in,
                              void* d_out, int out_size, void* d_ws, size_t ws_size,
                              hipStream_t stream) {
  (void)in_sizes; (void)n_in; (void)out_size; (void)ws_size;
  const float* x        = (const float*)d_in[0];
  const float* node_ts  = (const float*)d_in[1];
  const float* edge_ts  = (const float*)d_in[2];
  const float* W1  = (const float*)d_in[3];
  const float* b1  = (const float*)d_in[4];
  const float* g1  = (const float*)d_in[5];
  const float* be1 = (const float*)d_in[6];
  const float* rm1 = (const float*)d_in[7];
  const float* rv1 = (const float*)d_in[8];
  const float* W2  = (const float*)d_in[9];
  const float* b2  = (const float*)d_in[10];
  const float* g2  = (const float*)d_in[11];
  const float* be2 = (const float*)d_in[12];
  const float* rm2 = (const float*)d_in[13];
  const float* rv2 = (const float*)d_in[14];
  const int* batch    = (const int*)d_in[15];
  const int* edge_idx = (const int*)d_in[16];
  const int* degrees  = (const int*)d_in[17];
  float* out = (float*)d_out;

  // Workspace layout (all 256B aligned):
  //   [0)        cn   : M1 ints            (135168 B)
  //   [135168)   cts  : M1 floats          (135168 B)
  //   [270336)   h1   : M1*128 floats      (17301504 B)  -- agg1 then in-place GEMM
  //   [17571840) agg2 : B*128 floats       (524288 B)
  char* ws = (char*)d_ws;
  int*   cn   = (int*)ws;
  float* cts  = (float*)(ws + 135168);
  float* h1   = (float*)(ws + 270336);
  float* agg2 = (float*)(ws + 270336 + (size_t)M1 * CH * sizeof(float));

  build_cn_cts<<<M1 / 256, 256, 0, stream>>>(batch, edge_idx, edge_ts, node_ts,
                                             cn, cts);
  agg_layer1<<<M1 / 8, 256, 0, stream>>>(x, edge_ts, edge_idx, degrees,
                                         cn, cts, h1);
  gemm_bn_relu<<<M1 / 16, 256, 0, stream>>>(h1, W1, b1, g1, be1, rm1, rv1, h1);
  agg_layer2<<<BB / 8, 256, 0, stream>>>(h1, edge_ts, degrees, batch, node_ts,
                                         agg2);
  gemm_bn_relu<<<BB / 16, 256, 0, stream>>>(agg2, W2, b2, g2, be2, rm2, rv2, out);
}